// HGNN_48198122996183
// MI455X (gfx1250) — compile-verified
//
#include <hip/hip_runtime.h>
#include <hip/hip_bf16.h>

typedef __attribute__((ext_vector_type(2))) float v2f;
typedef __attribute__((ext_vector_type(8))) float v8f;

#define BN_EPS 1e-5f

// V_WMMA_F32_16X16X4_F32: D(16x16 f32) = A(16x4 f32) * B(4x16 f32) + C
// 8-arg pattern: (neg_a, A, neg_b, B, c_mod, C, reuse_a, reuse_b)
__device__ __forceinline__ v8f wmma4(v2f a, v2f b, v8f c) {
    return __builtin_amdgcn_wmma_f32_16x16x4_f32(
        /*neg_a=*/false, a, /*neg_b=*/false, b,
        /*c_mod=*/(short)0, c, /*reuse_a=*/false, /*reuse_b=*/false);
}

// ---------------------------------------------------------------------------
// Kernel 1: one block per batch element b.
//   h1 = relu(x[b] @ W1 + b1)              (512x16)@(16x32) -> LDS
//   h2 = relu(h1 @ W2 + b2)                (512x32)@(32x64) -> LDS
//   hagg[n,c] = max_l h2[l,c]*tree[n,l]    (128x64)         -> LDS (aliases h1)
//   m[b] = relu(hagg @ W3 + b3)            (128x64)@(64x16) -> global ws
// LDS: region0 = 64KB (h1, later hagg), region1 = 128KB (h2) => 192KB dynamic.
// ---------------------------------------------------------------------------
__global__ void __launch_bounds__(256) hgnn_stage123(
    const float* __restrict__ x, const float* __restrict__ tree,
    const float* __restrict__ W1, const float* __restrict__ b1,
    const float* __restrict__ W2, const float* __restrict__ b2,
    const float* __restrict__ W3, const float* __restrict__ b3,
    float* __restrict__ m_out)
{
    extern __shared__ float lds[];
    float* h1   = lds;           // 512*32 floats
    float* h2   = lds + 16384;   // 512*64 floats
    float* hagg = lds;           // 128*64 floats, reuses h1 region

    const int b    = blockIdx.x;
    const int tid  = threadIdx.x;
    const int lane = tid & 31;
    const int wave = tid >> 5;
    const int lo   = lane & 15;  // column / row-in-half
    const int hi   = lane >> 4;  // which lane-half

    const float* xb = x + b * (512 * 16);

    // ---- stage 1: h1 = relu(x @ W1 + b1); 32 row-tiles x 2 col-tiles ----
    for (int t = wave; t < 64; t += 8) {
        const int rb = (t >> 1) * 16;
        const int cb = (t & 1) * 16;
        v8f acc = {};
        for (int k0 = 0; k0 < 16; k0 += 4) {
            const int ka = k0 + hi * 2;
            v2f a, bm;
            a.x  = xb[(rb + lo) * 16 + ka];
            a.y  = xb[(rb + lo) * 16 + ka + 1];
            bm.x = W1[ka * 32 + cb + lo];
            bm.y = W1[(ka + 1) * 32 + cb + lo];
            acc = wmma4(a, bm, acc);
        }
        const float bias = b1[cb + lo];
        for (int i = 0; i < 8; ++i)
            h1[(rb + hi * 8 + i) * 32 + cb + lo] = fmaxf(acc[i] + bias, 0.0f);
    }
    __syncthreads();

    // ---- stage 2: h2 = relu(h1 @ W2 + b2); 32 row-tiles x 4 col-tiles ----
    for (int t = wave; t < 128; t += 8) {
        const int rb = (t >> 2) * 16;
        const int cb = (t & 3) * 16;
        v8f acc = {};
        for (int k0 = 0; k0 < 32; k0 += 4) {
            const int ka = k0 + hi * 2;
            v2f a, bm;
            a.x  = h1[(rb + lo) * 32 + ka];
            a.y  = h1[(rb + lo) * 32 + ka + 1];
            bm.x = W2[ka * 64 + cb + lo];
            bm.y = W2[(ka + 1) * 64 + cb + lo];
            acc = wmma4(a, bm, acc);
        }
        const float bias = b2[cb + lo];
        for (int i = 0; i < 8; ++i)
            h2[(rb + hi * 8 + i) * 64 + cb + lo] = fmaxf(acc[i] + bias, 0.0f);
    }
    __syncthreads();  // stage-2 reads of h1 done; hagg may overwrite it

    // ---- masked multiply-max aggregation: hagg[n,c] = max_l h2[l,c]*tree[n,l] ----
    for (int idx = tid; idx < 128 * 64; idx += 256) {
        const int n = idx >> 6;
        const int c = idx & 63;
        const float* trow = tree + n * 512;
        float v = -INFINITY;
        for (int l = 0; l < 512; ++l)
            v = fmaxf(v, h2[l * 64 + c] * trow[l]);
        hagg[idx] = v;  // idx == n*64 + c
    }
    __syncthreads();

    // ---- stage 3: m[b] = relu(hagg @ W3 + b3); 8 row-tiles, 1 col-tile ----
    {
        const int rb = wave * 16;  // 8 waves -> 8 tiles, exactly
        v8f acc = {};
        for (int k0 = 0; k0 < 64; k0 += 4) {
            const int ka = k0 + hi * 2;
            v2f a, bm;
            a.x  = hagg[(rb + lo) * 64 + ka];
            a.y  = hagg[(rb + lo) * 64 + ka + 1];
            bm.x = W3[ka * 16 + lo];
            bm.y = W3[(ka + 1) * 16 + lo];
            acc = wmma4(a, bm, acc);
        }
        const float bias = b3[lo];
        float* mb = m_out + b * 2048;
        for (int i = 0; i < 8; ++i)
            mb[(rb + hi * 8 + i) * 16 + lo] = fmaxf(acc[i] + bias, 0.0f);
    }
}

// ---------------------------------------------------------------------------
// Kernel 2: single block.
//   y = relu(m @ W4 + b4)   (32x2048)@(2048x64), 8 WMMA tiles (1/wave)
//   BatchNorm over batch (biased var), then out = yn @ W5 + b5 (32x64)@(64x10)
// ---------------------------------------------------------------------------
__global__ void __launch_bounds__(256) hgnn_stage45(
    const float* __restrict__ m, const float* __restrict__ W4,
    const float* __restrict__ b4, const float* __restrict__ gamma,
    const float* __restrict__ beta, const float* __restrict__ W5,
    const float* __restrict__ b5, float* __restrict__ out)
{
    __shared__ float y[32 * 64];
    __shared__ float scale[64];
    __shared__ float shift[64];

    const int tid  = threadIdx.x;
    const int lane = tid & 31;
    const int wave = tid >> 5;
    const int lo   = lane & 15;
    const int hi   = lane >> 4;

    // ---- stage 4: y = relu(m @ W4 + b4); 2 row-tiles x 4 col-tiles ----
    {
        const int rb = (wave >> 2) * 16;
        const int cb = (wave & 3) * 16;
        v8f acc = {};
        for (int k0 = 0; k0 < 2048; k0 += 4) {
            const int ka = k0 + hi * 2;
            v2f a, bm;
            a.x  = m[(rb + lo) * 2048 + ka];
            a.y  = m[(rb + lo) * 2048 + ka + 1];
            bm.x = W4[ka * 64 + cb + lo];
            bm.y = W4[(ka + 1) * 64 + cb + lo];
            acc = wmma4(a, bm, acc);
        }
        const float bias = b4[cb + lo];
        for (int i = 0; i < 8; ++i)
            y[(rb + hi * 8 + i) * 64 + cb + lo] = fmaxf(acc[i] + bias, 0.0f);
    }
    __syncthreads();

    // ---- BatchNorm (training-mode batch stats, biased variance) ----
    if (tid < 64) {
        float s = 0.0f;
        for (int r = 0; r < 32; ++r) s += y[r * 64 + tid];
        const float mean = s * (1.0f / 32.0f);
        float v = 0.0f;
        for (int r = 0; r < 32; ++r) {
            const float d = y[r * 64 + tid] - mean;
            v += d * d;
        }
        v *= (1.0f / 32.0f);
        const float sc = rsqrtf(v + BN_EPS) * gamma[tid];
        scale[tid] = sc;
        shift[tid] = beta[tid] - mean * sc;
    }
    __syncthreads();

    // ---- stage 5: out = yn @ W5 + b5; 320 outputs ----
    if (tid < 320) {
        const int r = tid / 10;
        const int o = tid % 10;
        float acc = b5[o];
        for (int c = 0; c < 64; ++c)
            acc += (y[r * 64 + c] * scale[c] + shift[c]) * W5[c * 10 + o];
        out[tid] = acc;
    }
}

extern "C" void kernel_launch(void* const* d_in, const int* in_sizes, int n_in,
                              void* d_out, int out_size, void* d_ws, size_t ws_size,
                              hipStream_t stream) {
    const float* x     = (const float*)d_in[0];
    const float* tree  = (const float*)d_in[1];
    const float* W1    = (const float*)d_in[2];
    const float* b1    = (const float*)d_in[3];
    const float* W2    = (const float*)d_in[4];
    const float* b2    = (const float*)d_in[5];
    const float* W3    = (const float*)d_in[6];
    const float* b3    = (const float*)d_in[7];
    const float* W4    = (const float*)d_in[8];
    const float* b4    = (const float*)d_in[9];
    const float* gamma = (const float*)d_in[10];
    const float* beta  = (const float*)d_in[11];
    const float* W5    = (const float*)d_in[12];
    const float* b5    = (const float*)d_in[13];

    float* m_ws = (float*)d_ws;      // 32*2048 floats = 256 KB scratch
    float* outp = (float*)d_out;     // 32*10 floats

    // 192 KB dynamic LDS (h1 64KB + h2 128KB; hagg aliases h1) — within 320KB/WGP.
    hgnn_stage123<<<32, 256, 196608, stream>>>(x, tree, W1, b1, W2, b2, W3, b3, m_ws);
    hgnn_stage45<<<1, 256, 0, stream>>>(m_ws, W4, b4, gamma, beta, W5, b5, outp);
}